// Multihead_Conv_25683904430309
// MI455X (gfx1250) — compile-verified
//
#include <hip/hip_runtime.h>
#include <stdint.h>

// Problem sizes (fixed by the reference).
#define B_SZ    32
#define L_SZ    4096
#define D_SZ    512
#define T_OUT   28                 // output rows (along L) per tile
#define ROWS    (T_OUT + 4)        // +2 halo each side (conv +-1, pool +-1 more)
#define DVEC    (D_SZ / 4)         // 128 float4 chunks per row
#define ROW_BYTES (D_SZ * 4)       // 2048 B: global row stride == LDS slot stride
#define BLOCK   128                // one thread per float4 chunk of D
#define N_TILES ((L_SZ + T_OUT - 1) / T_OUT)   // 147

typedef float f4 __attribute__((ext_vector_type(4)));

typedef int v4i __attribute__((ext_vector_type(4)));
typedef __attribute__((address_space(1))) v4i* g_v4i_p;   // global src
typedef __attribute__((address_space(3))) v4i* l_v4i_p;   // LDS dst

// Compile-time for-loop: passes integral constant JJ into the body functor.
template <int N> struct IC { static constexpr int value = N; };

template <int JJ, int END, class F>
__device__ __forceinline__ void static_for(F&& f) {
  if constexpr (JJ < END) {
    f(IC<JJ>{});
    static_for<JJ + 1, END>(f);
  }
}

// Async 16B copy global -> LDS (gfx1250 GLOBAL_LOAD_ASYNC_TO_LDS_B128).
// INST_OFFSET is applied to BOTH the global and LDS addresses, so one base
// pointer pair + per-row immediate covers the whole tile.
template <int IMM_OFF>
__device__ __forceinline__ void async_cp16(const f4* gbase, f4* lbase) {
  __builtin_amdgcn_global_load_async_to_lds_b128(
      (g_v4i_p)(uintptr_t)gbase,
      (l_v4i_p)(uint32_t)(uintptr_t)lbase,
      IMM_OFF, 0 /*cpol*/);
}

__device__ __forceinline__ f4 conv_relu(const f4& xm2, const f4& xm1, const f4& xc,
                                        const f4& w0, const f4& w1, const f4& w2,
                                        const f4& bb) {
  f4 y;
  y.x = fmaxf(0.f, fmaf(xm2.x, w0.x, fmaf(xm1.x, w1.x, fmaf(xc.x, w2.x, bb.x))));
  y.y = fmaxf(0.f, fmaf(xm2.y, w0.y, fmaf(xm1.y, w1.y, fmaf(xc.y, w2.y, bb.y))));
  y.z = fmaxf(0.f, fmaf(xm2.z, w0.z, fmaf(xm1.z, w1.z, fmaf(xc.z, w2.z, bb.z))));
  y.w = fmaxf(0.f, fmaf(xm2.w, w0.w, fmaf(xm1.w, w1.w, fmaf(xc.w, w2.w, bb.w))));
  return y;
}

__device__ __forceinline__ f4 max3(const f4& a, const f4& b, const f4& c) {
  f4 o;
  o.x = fmaxf(fmaxf(a.x, b.x), c.x);
  o.y = fmaxf(fmaxf(a.y, b.y), c.y);
  o.z = fmaxf(fmaxf(a.z, b.z), c.z);
  o.w = fmaxf(fmaxf(a.w, b.w), c.w);
  return o;
}

// Load per-dim 3-tap weights for dims d0..d0+3 and transpose into per-tap vectors.
__device__ __forceinline__ void load_weights(const float* w, const float* bias, int d0,
                                             f4& w0, f4& w1, f4& w2, f4& bb) {
  const f4* wv = (const f4*)(w + (size_t)d0 * 3);   // 48B*t -> 16B aligned
  const f4 wa = wv[0], wb = wv[1], wc = wv[2];
  w0 = (f4){wa.x, wa.w, wb.z, wc.y};
  w1 = (f4){wa.y, wb.x, wb.w, wc.z};
  w2 = (f4){wa.z, wb.y, wc.x, wc.w};
  bb = *(const f4*)(bias + d0);
}

// -------- Interior tiles: l0 in [28, 4060], all rows in-bounds, zero checks --------
__global__ __launch_bounds__(BLOCK) void conv_pool_interior(
    const float* __restrict__ x, const float* __restrict__ w,
    const float* __restrict__ bias, float* __restrict__ out)
{
  __shared__ f4 tile[ROWS][DVEC];    // 64 KB: per-thread private slots

  const int t  = threadIdx.x;
  const int bi = blockIdx.y;
  const int l0 = (blockIdx.x + 1) * T_OUT;
  const int d0 = t * 4;

  f4 w0, w1, w2, bb;
  load_weights(w, bias, d0, w0, w1, w2, bb);

  const f4* xbase = (const f4*)x + ((size_t)bi * L_SZ + (l0 - 2)) * DVEC + t;
  f4*       obase = (f4*)out     + ((size_t)bi * L_SZ + l0) * DVEC + t;
  f4*       lbase = &tile[0][t];

  static_for<0, ROWS>([&](auto jc) {
    constexpr int jj = decltype(jc)::value;
    async_cp16<jj * ROW_BYTES>(xbase, lbase);   // one VADDR pair, one LDS VGPR
  });

  f4 xm2 = (f4)0.f, xm1 = (f4)0.f;   // x[r-2], x[r-1]
  f4 y2  = (f4)0.f, y1  = (f4)0.f;   // y[r-3], y[r-2] (relu>=0 => 0 == -inf pad)

  static_for<0, ROWS>([&](auto jc) {
    constexpr int jj = decltype(jc)::value;
    // Async copies complete in order: outstanding <= 31-jj  =>  row jj landed.
    __builtin_amdgcn_s_wait_asynccnt(ROWS - 1 - jj);
    const f4 xc = tile[jj][t];                     // ds_load_b128 (imm offset)

    f4 y0 = (f4)0.f;
    if constexpr (jj >= 2) y0 = conv_relu(xm2, xm1, xc, w0, w1, w2, bb);

    if constexpr (jj >= 4)                         // out row l0 + jj - 4
      __builtin_nontemporal_store(max3(y2, y1, y0), obase + (jj - 4) * DVEC);

    xm2 = xm1; xm1 = xc;
    y2 = y1;   y1 = y0;
  });
}

// -------- Edge tiles: l0 = 0 and l0 = 4088, with bounds checks --------
__global__ __launch_bounds__(BLOCK) void conv_pool_edge(
    const float* __restrict__ x, const float* __restrict__ w,
    const float* __restrict__ bias, float* __restrict__ out)
{
  __shared__ f4 tile[ROWS][DVEC];

  const int t  = threadIdx.x;
  const int bi = blockIdx.y;
  const int l0 = (blockIdx.x == 0) ? 0 : (N_TILES - 1) * T_OUT;   // 0 or 4088
  const int d0 = t * 4;

  f4 w0, w1, w2, bb;
  load_weights(w, bias, d0, w0, w1, w2, bb);

  const f4* xbase = (const f4*)x + ((size_t)bi * L_SZ + (l0 - 2)) * DVEC + t;
  f4*       obase = (f4*)out     + ((size_t)bi * L_SZ + l0) * DVEC + t;
  f4*       lbase = &tile[0][t];

  static_for<0, ROWS>([&](auto jc) {
    constexpr int jj = decltype(jc)::value;
    const int r = l0 - 2 + jj;                  // uniform runtime check
    if (r >= 0 && r < L_SZ)
      async_cp16<jj * ROW_BYTES>(xbase, lbase);
  });
  __builtin_amdgcn_s_wait_asynccnt(0);   // skipped rows => progressive counts invalid

  f4 xm2 = (f4)0.f, xm1 = (f4)0.f;
  f4 y2  = (f4)0.f, y1  = (f4)0.f;

  #pragma unroll
  for (int jj = 0; jj < ROWS; ++jj) {
    const int r  = l0 - 2 + jj;
    const int yr = r - 1;

    f4 xc = (f4)0.f;
    if (r >= 0 && r < L_SZ) xc = tile[jj][t];

    f4 y0 = (f4)0.f;
    if (jj >= 2 && yr >= 0 && yr < L_SZ)
      y0 = conv_relu(xm2, xm1, xc, w0, w1, w2, bb);

    if (jj >= 4) {
      const int orr = r - 2;
      if (orr < L_SZ)
        obase[(jj - 4) * DVEC] = max3(y2, y1, y0);
    }

    xm2 = xm1; xm1 = xc;
    y2 = y1;   y1 = y0;
  }
}

extern "C" void kernel_launch(void* const* d_in, const int* in_sizes, int n_in,
                              void* d_out, int out_size, void* d_ws, size_t ws_size,
                              hipStream_t stream) {
  (void)in_sizes; (void)n_in; (void)out_size; (void)d_ws; (void)ws_size;
  const float* x = (const float*)d_in[0];
  const float* w = (const float*)d_in[1];
  const float* b = (const float*)d_in[2];
  float* out     = (float*)d_out;

  // Interior tiles: l0 = 28 .. 4060 (145 tiles); outputs rows 28..4087.
  dim3 gi(N_TILES - 2, B_SZ);
  hipLaunchKernelGGL(conv_pool_interior, gi, dim3(BLOCK), 0, stream, x, w, b, out);
  // Edge tiles: l0 = 0 (rows 0..27) and l0 = 4088 (rows 4088..4095, clipped).
  dim3 ge(2, B_SZ);
  hipLaunchKernelGGL(conv_pool_edge, ge, dim3(BLOCK), 0, stream, x, w, b, out);
}